// DeformableAttentionLayer_6176162972004
// MI455X (gfx1250) — compile-verified
//
#include <hip/hip_runtime.h>
#include <hip/hip_bf16.h>
#include <math.h>

// ---------------------------------------------------------------------------
// Problem constants (match reference)
// ---------------------------------------------------------------------------
#define BATCH 1024
#define INCH  256
#define OUTCH 256
#define NH    8
#define NPTS  16
#define DK    32          // OUT / NH
#define FH    128
#define FW    128

typedef __attribute__((ext_vector_type(2))) float v2f;
typedef __attribute__((ext_vector_type(8))) float v8f;

// ---------------------------------------------------------------------------
// Kernel 1/2/5: C[M x N] = A[M x 256] * W[N x 256]^T + bias[N]
// One wave32 per 16x16 output tile, V_WMMA_F32_16X16X4_F32, K-loop step 8.
// Two independent accumulators (even/odd K blocks) to break the WMMA->WMMA
// RAW chain and double matrix-pipe ILP at low occupancy.
//
// A-frag layout (16x4 f32): lanes 0-15 -> M=lane, K={k0,k0+1};
//                           lanes 16-31 -> M=lane-16, K={k0+2,k0+3}
// B-frag layout (4x16 f32): lanes 0-15 -> N=lane, K={k0,k0+1};
//                           lanes 16-31 -> N=lane-16, K={k0+2,k0+3}
// Since B[k][n] = W[n][k], both frags are single contiguous float2 loads.
// C/D layout (16x16 f32): VGPR r -> row (half*8 + r), col = lane&15.
// ---------------------------------------------------------------------------
__global__ __launch_bounds__(256)
void gemm_xwT_wmma_kernel(const float* __restrict__ A,
                          const float* __restrict__ W,
                          const float* __restrict__ bias,
                          float* __restrict__ C,
                          int M, int N)
{
    const int K = 256;
    const int wave  = (int)((blockIdx.x * blockDim.x + threadIdx.x) >> 5);
    const int lane  = (int)(threadIdx.x & 31);
    const int ntile = N >> 4;
    const int mt = wave / ntile;
    const int nt = wave - mt * ntile;
    if (mt >= (M >> 4)) return;

    const int half = lane >> 4;     // 0: K pair {0,1}; 1: K pair {2,3}
    const int l    = lane & 15;     // A: row m; B: col n

    const float* __restrict__ arow = A + (size_t)(mt * 16 + l) * K + 2 * half;
    const float* __restrict__ wrow = W + (size_t)(nt * 16 + l) * K + 2 * half;

    v8f acc0 = {};
    v8f acc1 = {};
#pragma unroll 4
    for (int k0 = 0; k0 < K; k0 += 8) {
        v2f a0 = *(const v2f*)(arow + k0);
        v2f b0 = *(const v2f*)(wrow + k0);
        v2f a1 = *(const v2f*)(arow + k0 + 4);
        v2f b1 = *(const v2f*)(wrow + k0 + 4);
        acc0 = __builtin_amdgcn_wmma_f32_16x16x4_f32(
                   false, a0, false, b0, (short)0, acc0, false, false);
        acc1 = __builtin_amdgcn_wmma_f32_16x16x4_f32(
                   false, a1, false, b1, (short)0, acc1, false, false);
    }

    const float bn = bias[nt * 16 + l];
    float* __restrict__ crow = C + (size_t)(mt * 16 + half * 8) * N + nt * 16 + l;
#pragma unroll
    for (int r = 0; r < 8; ++r)
        crow[(size_t)r * N] = acc0[r] + acc1[r] + bn;
}

// ---------------------------------------------------------------------------
// Kernel 3: ref = x*Wpos^T + bpos ; vel = x*Wvel^T + bvel   (B x 2 each)
// One thread per output element (4096 total); weights live in L2.
// ---------------------------------------------------------------------------
__global__ __launch_bounds__(256)
void refvel_kernel(const float* __restrict__ x,
                   const float* __restrict__ Wpos, const float* __restrict__ bpos,
                   const float* __restrict__ Wvel, const float* __restrict__ bvel,
                   float* __restrict__ ref, float* __restrict__ vel)
{
    const int tid = (int)(blockIdx.x * blockDim.x + threadIdx.x);
    if (tid >= BATCH * 4) return;
    const int b = tid >> 2;
    const int j = tid & 3;                        // 0,1 -> ref ; 2,3 -> vel
    const float* __restrict__ w  = (j < 2) ? (Wpos + (size_t)j * INCH)
                                           : (Wvel + (size_t)(j - 2) * INCH);
    const float* __restrict__ xr = x + (size_t)b * INCH;
    float s = 0.f;
#pragma unroll 8
    for (int k = 0; k < INCH; ++k) s = fmaf(xr[k], w[k], s);
    if (j < 2) ref[b * 2 + j]       = s + bpos[j];
    else       vel[b * 2 + (j - 2)] = s + bvel[j - 2];
}

// ---------------------------------------------------------------------------
// Kernel 4: bilinear deformable sampling + per-head softmax attention.
// One wave32 per (batch b, head h); lane = channel c (0..31).
//   v[p]   : sampled value for point p at this lane's channel (registers)
//   score  : cross-lane reduce of q[c]*v[p][c] via __shfl_xor (wave32)
//   softmax over the 16 points computed redundantly in every lane
//   out_h[b, h*32+c] written coalesced.
// ---------------------------------------------------------------------------
__device__ __forceinline__ float fm_corner(const float* __restrict__ fmc,
                                           int y, int x)
{
    const bool valid = (x >= 0) & (x < FW) & (y >= 0) & (y < FH);
    const int xi = min(max(x, 0), FW - 1);
    const int yi = min(max(y, 0), FH - 1);
    const float f = fmc[yi * FW + xi];
    return valid ? f : 0.f;
}

__global__ __launch_bounds__(256)
void sample_attn_kernel(const float* __restrict__ query,    // B x 256
                        const float* __restrict__ offsets,  // B x 256
                        const float* __restrict__ ref,      // B x 2
                        const float* __restrict__ fm,       // B x 32 x 128 x 128
                        float* __restrict__ out_h)          // B x 256
{
    const int wave = (int)((blockIdx.x * blockDim.x + threadIdx.x) >> 5);
    const int lane = (int)(threadIdx.x & 31);                // channel c
    if (wave >= BATCH * NH) return;
    const int b = wave >> 3;
    const int h = wave & 7;

    const float q  = query[(size_t)b * OUTCH + h * DK + lane];
    const float rx = ref[b * 2 + 0];
    const float ry = ref[b * 2 + 1];
    const float* __restrict__ off = offsets + (size_t)b * OUTCH + h * (NPTS * 2);
    const float* __restrict__ fmc = fm + ((size_t)b * DK + lane) * (FH * FW);

    float v[NPTS];
    float sc[NPTS];

#pragma unroll
    for (int p = 0; p < NPTS; ++p) {
        const float gx = rx + off[2 * p + 0];
        const float gy = ry + off[2 * p + 1];
        // grid_sample, align_corners=False
        const float ixf = ((gx + 1.f) * (float)FW - 1.f) * 0.5f;
        const float iyf = ((gy + 1.f) * (float)FH - 1.f) * 0.5f;
        const float x0f = floorf(ixf), y0f = floorf(iyf);
        const int   x0  = (int)x0f,    y0  = (int)y0f;
        const int   x1  = x0 + 1,      y1  = y0 + 1;
        const float wx1 = ixf - x0f,   wx0 = 1.f - wx1;
        const float wy1 = iyf - y0f,   wy0 = 1.f - wy1;

        float val = fm_corner(fmc, y0, x0) * (wy0 * wx0)
                  + fm_corner(fmc, y0, x1) * (wy0 * wx1)
                  + fm_corner(fmc, y1, x0) * (wy1 * wx0)
                  + fm_corner(fmc, y1, x1) * (wy1 * wx1);
        v[p] = val;

        // wave32 reduction of q[c] * v[p][c] over the 32 channels
        float t = q * val;
#pragma unroll
        for (int s = 16; s >= 1; s >>= 1)
            t += __shfl_xor(t, s, 32);
        sc[p] = t * 0.17677669529663687f;   // / sqrt(32)
    }

    // softmax over the 16 points (identical in every lane)
    float mx = sc[0];
#pragma unroll
    for (int p = 1; p < NPTS; ++p) mx = fmaxf(mx, sc[p]);
    float e[NPTS];
    float sum = 0.f;
#pragma unroll
    for (int p = 0; p < NPTS; ++p) { e[p] = __expf(sc[p] - mx); sum += e[p]; }
    const float inv = 1.f / sum;

    float o = 0.f;
#pragma unroll
    for (int p = 0; p < NPTS; ++p) o = fmaf(e[p] * inv, v[p], o);

    out_h[(size_t)b * OUTCH + h * DK + lane] = o;
}

// ---------------------------------------------------------------------------
// Launch: GEMM(query) -> GEMM(offsets) -> ref/vel -> sample+attn -> GEMM(out)
// ---------------------------------------------------------------------------
extern "C" void kernel_launch(void* const* d_in, const int* in_sizes, int n_in,
                              void* d_out, int out_size, void* d_ws, size_t ws_size,
                              hipStream_t stream)
{
    const float* x    = (const float*)d_in[0];
    const float* fm   = (const float*)d_in[1];
    const float* Wq   = (const float*)d_in[2];
    const float* bq   = (const float*)d_in[3];
    const float* Woff = (const float*)d_in[4];
    const float* boff = (const float*)d_in[5];
    const float* Wpos = (const float*)d_in[6];
    const float* bpos = (const float*)d_in[7];
    const float* Wvel = (const float*)d_in[8];
    const float* bvel = (const float*)d_in[9];
    const float* Wout = (const float*)d_in[10];
    const float* bout = (const float*)d_in[11];

    float* out = (float*)d_out;                    // (B, 256)
    float* ref = out + (size_t)BATCH * OUTCH;      // (B, 2)
    float* vel = ref + (size_t)BATCH * 2;          // (B, 2)

    float* ws    = (float*)d_ws;
    float* query = ws;                             // (B, 256)
    float* offs  = ws + (size_t)BATCH * OUTCH;     // (B, 256)
    float* outh  = ws + (size_t)2 * BATCH * OUTCH; // (B, 256)

    // WMMA GEMM config: (M/16)*(N/16) waves, 8 waves (256 threads) per block
    const int gemm_waves  = (BATCH / 16) * (OUTCH / 16);   // 1024
    const int gemm_blocks = gemm_waves / 8;                // 128

    // 1) query = x @ Wq^T + bq
    gemm_xwT_wmma_kernel<<<gemm_blocks, 256, 0, stream>>>(
        x, Wq, bq, query, BATCH, OUTCH);

    // 2) offsets = x @ Woff^T + boff   (NH*NPTS*2 == 256 == OUTCH)
    gemm_xwT_wmma_kernel<<<gemm_blocks, 256, 0, stream>>>(
        x, Woff, boff, offs, BATCH, OUTCH);

    // 3) ref / vel
    refvel_kernel<<<(BATCH * 4 + 255) / 256, 256, 0, stream>>>(
        x, Wpos, bpos, Wvel, bvel, ref, vel);

    // 4) deformable sampling + attention -> out_h
    const int attn_waves  = BATCH * NH;                    // 8192
    const int attn_blocks = attn_waves / 8;                // 1024
    sample_attn_kernel<<<attn_blocks, 256, 0, stream>>>(
        query, offs, ref, fm, outh);

    // 5) output = out_h @ Wout^T + bout
    gemm_xwT_wmma_kernel<<<gemm_blocks, 256, 0, stream>>>(
        outh, Wout, bout, out, BATCH, OUTCH);
}